// Nms3dAxial_63737314673341
// MI455X (gfx1250) — compile-verified
//
#include <hip/hip_runtime.h>
#include <hip/hip_bf16.h>
#include <stdint.h>

// results row layout: [score, cls, cx, cy, cz, sx, sy, sz]
//
// Workspace layout (u64 "slots", B = gridDim of the pass kernels):
//   [0*B .. 1*B) pass1 per-block best, class 0   (rank-0 candidates)
//   [1*B .. 2*B) pass1 per-block best, class 1
//   [2*B .. 3*B) pass2 per-block best, class 0   (rank-1 candidates)
//   [3*B .. 4*B) pass2 per-block best, class 1
// Every block overwrites its slots unconditionally each launch -> no init pass,
// no atomics, no stale state across graph replays.
//
// pack = (score_bits << 32) | (0xFFFFFFFF - index):
//   positive-float bit pattern is order-preserving, ~index makes score ties
//   resolve to the LOWER index — exactly matching JAX's stable argsort order.

#define NMS_THREADS 256
#define NMS_WAVES   (NMS_THREADS / 32)

static __device__ __forceinline__ unsigned long long pack_cand(float score, unsigned idx) {
  return ((unsigned long long)__float_as_uint(score) << 32) |
         (unsigned long long)(0xFFFFFFFFu - idx);
}

static __device__ __forceinline__ unsigned long long wave_max_u64(unsigned long long v) {
  // wave32 butterfly reduction (CDNA5 is wave32-only; width must be 32, not 64)
#pragma unroll
  for (int off = 16; off > 0; off >>= 1) {
    unsigned long long o = __shfl_xor(v, off, 32);
    v = (o > v) ? o : v;
  }
  return v;
}

// ---------------------------------------------------------------------------
// Pass 1: per-class best (score, index). Streams only the first 8 bytes of
// each row (global_load_b64). Block result written to private slot.
// ---------------------------------------------------------------------------
__global__ __launch_bounds__(NMS_THREADS)
void nms_pass1(const float* __restrict__ results, int n,
               unsigned long long* __restrict__ slots, int B) {
  __shared__ unsigned long long red[2][NMS_WAVES];

  unsigned long long b0 = 0ull, b1 = 0ull;
  int stride = gridDim.x * blockDim.x;
  for (int i = blockIdx.x * blockDim.x + threadIdx.x; i < n; i += stride) {
    float2 sc = *reinterpret_cast<const float2*>(results + (size_t)i * 8);
    if (sc.x >= 0.0f) {  // SCORE_TR == 0.0
      unsigned long long p = pack_cand(sc.x, (unsigned)i);
      if (sc.y == 0.0f) { b0 = (p > b0) ? p : b0; }
      else              { b1 = (p > b1) ? p : b1; }
    }
  }
  b0 = wave_max_u64(b0);
  b1 = wave_max_u64(b1);
  int wave = threadIdx.x >> 5, lane = threadIdx.x & 31;
  if (lane == 0) { red[0][wave] = b0; red[1][wave] = b1; }
  __syncthreads();
  if (threadIdx.x == 0) {
    unsigned long long r0 = 0ull, r1 = 0ull;
#pragma unroll
    for (int k = 0; k < NMS_WAVES; ++k) {
      r0 = (red[0][k] > r0) ? red[0][k] : r0;
      r1 = (red[1][k] > r1) ? red[1][k] : r1;
    }
    slots[blockIdx.x]     = r0;  // unconditional overwrite
    slots[B + blockIdx.x] = r1;
  }
}

// ---------------------------------------------------------------------------
// Pass 2: per-class best among boxes with IoU(rank-0 winner, box) <= 0.3.
// First re-reduces pass1's tiny slot array (L2-resident) to recover the
// winners, then streams all rows. Rows are staged through LDS with the
// gfx1250 async copy engine:
//   global_load_async_to_lds_b128 (x2; offset:16 applies to both LDS and
//   global side per the ISA addressing formula), s_wait_asynccnt, then
//   ds_load_b128 readback.
// ---------------------------------------------------------------------------
__global__ __launch_bounds__(NMS_THREADS)
void nms_pass2(const float* __restrict__ results, int n,
               unsigned long long* __restrict__ slots, int B) {
  __shared__ float4 stage[NMS_THREADS * 2];  // 32 B per thread
  __shared__ unsigned long long red[2][NMS_WAVES];

  // --- block-wide reduction of pass1 slots -> rank-0 winners (all threads) ---
  unsigned long long l0 = 0ull, l1 = 0ull;
  for (int t = threadIdx.x; t < B; t += blockDim.x) {
    unsigned long long v0 = slots[t], v1 = slots[B + t];
    l0 = (v0 > l0) ? v0 : l0;
    l1 = (v1 > l1) ? v1 : l1;
  }
  l0 = wave_max_u64(l0);
  l1 = wave_max_u64(l1);
  int wave = threadIdx.x >> 5, lane = threadIdx.x & 31;
  if (lane == 0) { red[0][wave] = l0; red[1][wave] = l1; }
  __syncthreads();
  unsigned long long w0 = 0ull, w1 = 0ull;
#pragma unroll
  for (int k = 0; k < NMS_WAVES; ++k) {
    w0 = (red[0][k] > w0) ? red[0][k] : w0;
    w1 = (red[1][k] > w1) ? red[1][k] : w1;
  }

  // --- winner boxes (broadcast loads, L2/WGP$ hits) ---
  unsigned widx0 = 0xFFFFFFFFu - (unsigned)(w0 & 0xFFFFFFFFull);
  unsigned widx1 = 0xFFFFFFFFu - (unsigned)(w1 & 0xFFFFFFFFull);
  bool has0 = (w0 != 0ull), has1 = (w1 != 0ull);
  float mn0x=0.f,mn0y=0.f,mn0z=0.f, mx0x=0.f,mx0y=0.f,mx0z=0.f, va0=0.f;
  float mn1x=0.f,mn1y=0.f,mn1z=0.f, mx1x=0.f,mx1y=0.f,mx1z=0.f, va1=0.f;
  if (has0) {
    const float* r = results + (size_t)widx0 * 8;
    float cx=r[2],cy=r[3],cz=r[4],sx=r[5],sy=r[6],sz=r[7];
    mn0x=cx-0.5f*sx; mn0y=cy-0.5f*sy; mn0z=cz-0.5f*sz;
    mx0x=cx+0.5f*sx; mx0y=cy+0.5f*sy; mx0z=cz+0.5f*sz;
    va0 = sx*sy*sz;
  }
  if (has1) {
    const float* r = results + (size_t)widx1 * 8;
    float cx=r[2],cy=r[3],cz=r[4],sx=r[5],sy=r[6],sz=r[7];
    mn1x=cx-0.5f*sx; mn1y=cy-0.5f*sy; mn1z=cz-0.5f*sz;
    mx1x=cx+0.5f*sx; mx1y=cy+0.5f*sy; mx1z=cz+0.5f*sz;
    va1 = sx*sy*sz;
  }

  const unsigned lds_addr = (unsigned)(uintptr_t)(&stage[threadIdx.x * 2]);

  // --- main stream: IoU against the per-class winner ---
  unsigned long long b0 = 0ull, b1 = 0ull;
  int stride = gridDim.x * blockDim.x;
  for (int i = blockIdx.x * blockDim.x + threadIdx.x; i < n; i += stride) {
    unsigned long long gaddr = (unsigned long long)(uintptr_t)(results + (size_t)i * 8);
    asm volatile(
        "global_load_async_to_lds_b128 %0, %1, off\n\t"
        "global_load_async_to_lds_b128 %0, %1, off offset:16"
        :
        : "v"(lds_addr), "v"(gaddr)
        : "memory");
    asm volatile("s_wait_asynccnt 0x0" ::: "memory");

    float4 lo = stage[threadIdx.x * 2];      // score, cls, cx, cy
    float4 hi = stage[threadIdx.x * 2 + 1];  // cz, sx, sy, sz

    int c = (lo.y == 0.0f) ? 0 : 1;
    bool has   = c ? has1  : has0;
    unsigned w = c ? widx1 : widx0;
    if (lo.x >= 0.0f && has && (unsigned)i != w) {
      float amnx = c ? mn1x : mn0x, amny = c ? mn1y : mn0y, amnz = c ? mn1z : mn0z;
      float amxx = c ? mx1x : mx0x, amxy = c ? mx1y : mx0y, amxz = c ? mx1z : mx0z;
      float va   = c ? va1  : va0;

      float bmnx = lo.z - 0.5f*hi.y, bmxx = lo.z + 0.5f*hi.y;
      float bmny = lo.w - 0.5f*hi.z, bmxy = lo.w + 0.5f*hi.z;
      float bmnz = hi.x - 0.5f*hi.w, bmxz = hi.x + 0.5f*hi.w;

      float ix = fmaxf(fminf(amxx, bmxx) - fmaxf(amnx, bmnx), 0.0f);
      float iy = fmaxf(fminf(amxy, bmxy) - fmaxf(amny, bmny), 0.0f);
      float iz = fmaxf(fminf(amxz, bmxz) - fmaxf(amnz, bmnz), 0.0f);
      float inter = ix * iy * iz;
      float vb = hi.y * hi.z * hi.w;
      float iou = inter / (va + vb - inter + 1e-7f);
      if (iou <= 0.3f) {
        unsigned long long p = pack_cand(lo.x, (unsigned)i);
        if (c == 0) { b0 = (p > b0) ? p : b0; }
        else        { b1 = (p > b1) ? p : b1; }
      }
    }
  }

  // --- block result -> private slots (WAR barrier before reusing `red`) ---
  b0 = wave_max_u64(b0);
  b1 = wave_max_u64(b1);
  __syncthreads();
  if (lane == 0) { red[0][wave] = b0; red[1][wave] = b1; }
  __syncthreads();
  if (threadIdx.x == 0) {
    unsigned long long r0 = 0ull, r1 = 0ull;
#pragma unroll
    for (int k = 0; k < NMS_WAVES; ++k) {
      r0 = (red[0][k] > r0) ? red[0][k] : r0;
      r1 = (red[1][k] > r1) ? red[1][k] : r1;
    }
    slots[2 * B + blockIdx.x] = r0;  // unconditional overwrite
    slots[3 * B + blockIdx.x] = r1;
  }
}

// ---------------------------------------------------------------------------
// Finalize: reduce all 4 slot arrays, emit 4 rows x 8 cols:
// [c0 rank0, c0 rank1, c1 rank0, c1 rank1], clamp >= 0, zeros if missing.
// ---------------------------------------------------------------------------
__global__ __launch_bounds__(NMS_THREADS)
void nms_finalize(const float* __restrict__ results,
                  const unsigned long long* __restrict__ slots, int B,
                  float* __restrict__ out) {
  __shared__ unsigned long long red[4][NMS_WAVES];

  unsigned long long l[4] = {0ull, 0ull, 0ull, 0ull};
  for (int t = threadIdx.x; t < B; t += blockDim.x) {
#pragma unroll
    for (int c = 0; c < 4; ++c) {
      unsigned long long v = slots[c * B + t];
      l[c] = (v > l[c]) ? v : l[c];
    }
  }
#pragma unroll
  for (int c = 0; c < 4; ++c) l[c] = wave_max_u64(l[c]);
  int wave = threadIdx.x >> 5, lane = threadIdx.x & 31;
  if (lane == 0) {
#pragma unroll
    for (int c = 0; c < 4; ++c) red[c][wave] = l[c];
  }
  __syncthreads();

  int t = threadIdx.x;
  if (t < 32) {
    int row = t >> 3, col = t & 7;
    // output rows 0..3 -> reduced arrays {0: c0r0, 2: c0r1, 1: c1r0, 3: c1r1}
    int cell = (row >> 1) + (row & 1) * 2;
    unsigned long long p = 0ull;
#pragma unroll
    for (int k = 0; k < NMS_WAVES; ++k) p = (red[cell][k] > p) ? red[cell][k] : p;
    float v = 0.0f;
    if (p != 0ull) {
      unsigned idx = 0xFFFFFFFFu - (unsigned)(p & 0xFFFFFFFFull);
      v = fmaxf(results[(size_t)idx * 8 + col], 0.0f);
    }
    out[t] = v;
  }
}

extern "C" void kernel_launch(void* const* d_in, const int* in_sizes, int n_in,
                              void* d_out, int out_size, void* d_ws, size_t ws_size,
                              hipStream_t stream) {
  const float* results = (const float*)d_in[0];
  int n = in_sizes[0] / 8;  // 20000 rows of 8 floats
  unsigned long long* slots = (unsigned long long*)d_ws;

  // 4 u64 slots per block; clamp grid to workspace capacity (grid-stride loops
  // keep any B >= 1 correct).
  int B = (n + NMS_THREADS - 1) / NMS_THREADS;
  long long bmax = (long long)(ws_size / (4 * sizeof(unsigned long long)));
  if (bmax < 1) bmax = 1;
  if ((long long)B > bmax) B = (int)bmax;
  if (B < 1) B = 1;

  nms_pass1<<<B, NMS_THREADS, 0, stream>>>(results, n, slots, B);
  nms_pass2<<<B, NMS_THREADS, 0, stream>>>(results, n, slots, B);
  nms_finalize<<<1, NMS_THREADS, 0, stream>>>(results, slots, B, (float*)d_out);
}